// RF_Pool_69114613730551
// MI455X (gfx1250) — compile-verified
//
#include <hip/hip_runtime.h>
#include <hip/hip_bf16.h>

// Problem constants (match reference): B=4, C=32, H=64, W=64, K=64, block 2x2
constexpr int Bn = 4, Cn = 32, Hn = 64, Wn = 64, Kn = 64;
constexpr int HW   = Hn * Wn;        // 4096 pixels per (b,c) image / per RF
constexpr int BDIM = 256;            // 8 wave32 waves per workgroup
constexpr int NW   = BDIM / 32;      // waves per workgroup
constexpr int PPT  = HW / BDIM;      // 16 pixels per thread
constexpr int VEC  = 4;              // float4 per async b128
constexpr int ISS  = HW / (BDIM * VEC); // async instructions per wave per tile = 4

#ifndef __has_builtin
#define __has_builtin(x) 0
#endif

#if __has_builtin(__builtin_amdgcn_global_load_async_to_lds_b128)
#define HAVE_ASYNC_LDS 1
#else
#define HAVE_ASYNC_LDS 0
#endif

#if __has_builtin(__builtin_amdgcn_s_wait_asynccnt)
#define WAIT_ASYNC(N) __builtin_amdgcn_s_wait_asynccnt(N)
#else
#define WAIT_ASYNC(N) asm volatile("s_wait_asynccnt %0" ::"n"(N) : "memory")
#endif

// Builtin operand type: pointer to 4 x i32 vector (b128 payload).
typedef int v4i __attribute__((ext_vector_type(4)));

// Kick off an async (ASYNCcnt-tracked) copy of one 16KB RF tile into LDS.
// Each of the 8 waves issues ISS=4 global_load_async_to_lds_b128 instructions
// (32 lanes x 16B = 512B per instruction), so per-wave ASYNCcnt rises by 4.
__device__ __forceinline__ void issue_rf_tile(const float* __restrict__ rfs,
                                              int k, float* dst, int tid) {
  const float* src = rfs + k * HW;
#if HAVE_ASYNC_LDS
#pragma unroll
  for (int j = 0; j < ISS; ++j) {
    const int idx = (j * BDIM + tid) * VEC;
    __builtin_amdgcn_global_load_async_to_lds_b128(
        (__attribute__((address_space(1))) v4i*)(src + idx),
        (__attribute__((address_space(3))) v4i*)(dst + idx),
        /*offset=*/0, /*cpol=*/0);
  }
#else
#pragma unroll
  for (int j = 0; j < ISS; ++j) {
    const int idx = (j * BDIM + tid) * VEC;
    *(float4*)(dst + idx) = *(const float4*)(src + idx);
  }
#endif
}

__global__ __launch_bounds__(BDIM) void rf_pool_prob_kernel(
    const float* __restrict__ u,    // (B, C, H, W)
    const float* __restrict__ rfs,  // (K, H, W)
    float* __restrict__ out) {      // (B, C, H/2, W/2)
  // LDS: double-buffered RF tile (32KB) + tiny cross-wave reduction scratch.
  __shared__ float rf_lds[2 * HW];
  __shared__ float redmax[NW];
  __shared__ float redsum[NW];

  const int tid = threadIdx.x;
  const int bc  = blockIdx.x;                 // one workgroup per (b,c)
  const float* ub = u + (size_t)bc * HW;

  // u tile and h accumulator live entirely in VGPRs (16 pixels per thread,
  // strided by BDIM for coalesced global loads and conflict-free LDS).
  float u_r[PPT], h_r[PPT];
#pragma unroll
  for (int t = 0; t < PPT; ++t) {
    u_r[t] = ub[t * BDIM + tid];
    h_r[t] = 0.0f;
  }

  // Prime the async pipeline with RF tile 0.
  issue_rf_tile(rfs, 0, rf_lds, tid);

  for (int k = 0; k < Kn; ++k) {
    float* cur = rf_lds + (k & 1) * HW;

    // Prefetch next tile, then retire only the current tile's 4 async ops
    // (async loads complete in order -> waiting to <=ISS leaves the prefetch
    // in flight behind the compute below).
    if (k + 1 < Kn) {
      issue_rf_tile(rfs, k + 1, rf_lds + ((k + 1) & 1) * HW, tid);
      WAIT_ASYNC(ISS);
    } else {
      WAIT_ASYNC(0);
    }
    __syncthreads();  // all waves' async data for `cur` visible

    // Pass 1: masked per-thread max of v = u*rf (off-state clamps at 0,
    // so initializing lmax=0 folds in reference's max(.,0)).
    float rf_r[PPT];
    float lmax = 0.0f;
#pragma unroll
    for (int t = 0; t < PPT; ++t) {
      const float r = cur[t * BDIM + tid];
      rf_r[t] = r;
      const float v = u_r[t] * r;
      lmax = (r > 0.0f) ? fmaxf(lmax, v) : lmax;
    }
    // wave32 tree reduce + cross-wave combine.
#pragma unroll
    for (int off = 16; off > 0; off >>= 1)
      lmax = fmaxf(lmax, __shfl_xor(lmax, off, 32));
    if ((tid & 31) == 0) redmax[tid >> 5] = lmax;
    __syncthreads();
    float m = 0.0f;
#pragma unroll
    for (int w = 0; w < NW; ++w) m = fmaxf(m, redmax[w]);

    // Pass 2: e = exp(v - m) on RF support; per-thread partial sum.
    float e_r[PPT];
    float lsum = 0.0f;
#pragma unroll
    for (int t = 0; t < PPT; ++t) {
      const float e = (rf_r[t] > 0.0f) ? __expf(u_r[t] * rf_r[t] - m) : 0.0f;
      e_r[t] = e;
      lsum += e;
    }
#pragma unroll
    for (int off = 16; off > 0; off >>= 1) lsum += __shfl_xor(lsum, off, 32);
    if ((tid & 31) == 0) redsum[tid >> 5] = lsum;
    __syncthreads();
    float tot = 0.0f;
#pragma unroll
    for (int w = 0; w < NW; ++w) tot += redsum[w];

    const float inv = 1.0f / (__expf(-m) + tot);  // denom includes off-state
#pragma unroll
    for (int t = 0; t < PPT; ++t) h_r[t] = fmaf(e_r[t], inv, h_r[t]);
    // No trailing barrier needed: next iteration's redmax write / buffer
    // overwrite are ordered after this iteration's second barrier.
  }

  // Stage h into (now idle) LDS buffer 0 and do the 2x2 block max-pool.
  float* h_s = rf_lds;
#pragma unroll
  for (int t = 0; t < PPT; ++t) h_s[t * BDIM + tid] = h_r[t];
  __syncthreads();

  float* ob = out + (size_t)bc * (Hn / 2) * (Wn / 2);
#pragma unroll
  for (int q = 0; q < (Hn / 2) * (Wn / 2) / BDIM; ++q) {  // 4 outputs/thread
    const int o = q * BDIM + tid;          // 0..1023 within (b,c)
    const int i = o >> 5, j = o & 31;      // 32x32 output coords
    const int p = (2 * i) * Wn + 2 * j;
    ob[o] = fmaxf(fmaxf(h_s[p], h_s[p + 1]),
                  fmaxf(h_s[p + Wn], h_s[p + Wn + 1]));
  }
}

extern "C" void kernel_launch(void* const* d_in, const int* in_sizes, int n_in,
                              void* d_out, int out_size, void* d_ws, size_t ws_size,
                              hipStream_t stream) {
  const float* u   = (const float*)d_in[0];  // (4,32,64,64) f32
  const float* rfs = (const float*)d_in[1];  // (64,64,64)   f32
  float* out = (float*)d_out;                // (4,32,32,32) f32
  (void)in_sizes; (void)n_in; (void)out_size; (void)d_ws; (void)ws_size;

  rf_pool_prob_kernel<<<Bn * Cn, BDIM, 0, stream>>>(u, rfs, out);
}